// SOMFNN_10273561772221
// MI455X (gfx1250) — compile-verified
//
#include <hip/hip_runtime.h>
#include <math.h>

typedef float v2f __attribute__((ext_vector_type(2)));
typedef float v4f __attribute__((ext_vector_type(4)));
typedef float v8f __attribute__((ext_vector_type(8)));
typedef int   v4i __attribute__((vector_size(4 * sizeof(int))));

#define AS1 __attribute__((address_space(1)))
#define AS3 __attribute__((address_space(3)))

#define DELTA_F 0.13533528f  /* float(np.exp(-2.0)) rounded to f32 */
#define MAXN 2048

#if defined(__has_builtin)
#  if __has_builtin(__builtin_amdgcn_global_load_async_to_lds_b128) && \
      __has_builtin(__builtin_amdgcn_s_wait_asynccnt)
#    define ASYNC_OK 1
#  else
#    define ASYNC_OK 0
#  endif
#else
#  define ASYNC_OK 0
#endif

// 16B global -> LDS copy (per lane). Async path on gfx1250 (ASYNCcnt).
__device__ __forceinline__ void g2lds_b128(const float* g, float* l) {
#if ASYNC_OK
    __builtin_amdgcn_global_load_async_to_lds_b128(
        (AS1 v4i*)(AS1 float*)g, (AS3 v4i*)(AS3 float*)l, 0, 0);
#else
    const AS1 v4f* gs = (const AS1 v4f*)(const AS1 float*)g;
    AS3 v4f* ls = (AS3 v4f*)(AS3 float*)l;
    *ls = *gs;
#endif
}

__device__ __forceinline__ void wait_async() {
#if ASYNC_OK
    __builtin_amdgcn_s_wait_asynccnt(0);
#endif
}

// ---------------------------------------------------------------------------
// Row squared norms: SEN[row] = sum_j X[row,j]^2
// ---------------------------------------------------------------------------
__global__ __launch_bounds__(256) void row_sumsq(const float* __restrict__ X,
                                                 float* __restrict__ SEN, int M) {
    __shared__ float red[256];
    const int row = blockIdx.x;
    const float* xr = X + (size_t)row * M;
    float s = 0.f;
    for (int j = threadIdx.x; j < M; j += blockDim.x) { float v = xr[j]; s += v * v; }
    red[threadIdx.x] = s;
    __syncthreads();
    for (int off = blockDim.x >> 1; off > 0; off >>= 1) {
        if (threadIdx.x < off) red[threadIdx.x] += red[threadIdx.x + off];
        __syncthreads();
    }
    if (threadIdx.x == 0) SEN[row] = red[0];
}

__global__ void zero_scalar(float* p) { *p = 0.f; }

// ---------------------------------------------------------------------------
// Accumulate sum_j (mean_i X[i,j])^2  into *gsq (atomic)
// ---------------------------------------------------------------------------
__global__ __launch_bounds__(256) void colmean_sq(const float* __restrict__ X,
                                                  float* __restrict__ gsq,
                                                  int N, int M) {
    __shared__ float red[256];
    const int j = blockIdx.x * blockDim.x + threadIdx.x;
    float s = 0.f;
    if (j < M) {
        for (int i = 0; i < N; ++i) s += X[(size_t)i * M + j];
    }
    float m = s / (float)N;
    red[threadIdx.x] = (j < M) ? m * m : 0.f;
    __syncthreads();
    for (int off = blockDim.x >> 1; off > 0; off >>= 1) {
        if (threadIdx.x < off) red[threadIdx.x] += red[threadIdx.x + off];
        __syncthreads();
    }
    if (threadIdx.x == 0) atomicAdd(gsq, red[0]);
}

// base = mean(SEN) - gsq
__global__ __launch_bounds__(256) void finalize_base(const float* __restrict__ SEN,
                                                     const float* __restrict__ gsq,
                                                     float* __restrict__ base, int N) {
    __shared__ float red[256];
    float s = 0.f;
    for (int i = threadIdx.x; i < N; i += blockDim.x) s += SEN[i];
    red[threadIdx.x] = s;
    __syncthreads();
    for (int off = blockDim.x >> 1; off > 0; off >>= 1) {
        if (threadIdx.x < off) red[threadIdx.x] += red[threadIdx.x + off];
        __syncthreads();
    }
    if (threadIdx.x == 0) *base = red[0] / (float)N - *gsq;
}

// ---------------------------------------------------------------------------
// WMMA f32 GEMM:  C[M,N] = A[M,K] * B[N,K]^T   (+ bias, sigmoid if FUSE)
// Block = 8 wave32 -> 128x64 macro-tile. K consumed in 32-wide chunks,
// double-buffered in LDS via async global->LDS B128 copies (ASYNCcnt).
// Fragment reads are conflict-free ds_load_b64 (row pitch 36 dwords).
// Per k-step: wave does 1 A-frag, 4 B-frags, 4 x V_WMMA_F32_16X16X4_F32.
// ---------------------------------------------------------------------------
template <bool FUSE>
__global__ __launch_bounds__(256) void wmma_gemm_xyT(const float* __restrict__ A,
                                                     const float* __restrict__ B,
                                                     const float* __restrict__ bias,
                                                     float* __restrict__ C,
                                                     int M, int N, int K) {
    constexpr int BM = 128, BN = 64, BK = 32, LDP = 36;
    __shared__ float As[2][BM][LDP];   // 2 x 18KB
    __shared__ float Bs[2][BN][LDP];   // 2 x  9KB

    const int lane = threadIdx.x;                 // 0..31 (wave32)
    const int wave = threadIdx.y;                 // 0..7
    const int t    = wave * 32 + lane;            // 0..255
    const int nbn  = N / BN;
    const int bm   = blockIdx.x / nbn;
    const int bn   = blockIdx.x % nbn;
    const int m0   = bm * BM, n0 = bn * BN;
    const int half = lane >> 4;
    const int l16  = lane & 15;

    auto copy_chunk = [&](int k0, int b) {
        // A tile: 128 rows x 32 floats = 1024 B128 chunks, 4 per thread
#pragma unroll
        for (int i = 0; i < 4; ++i) {
            const int c = t + i * 256;
            const int row = c >> 3, seg = c & 7;
            g2lds_b128(A + (size_t)(m0 + row) * K + k0 + seg * 4,
                       &As[b][row][seg * 4]);
        }
        // B tile: 64 rows x 32 floats = 512 B128 chunks, 2 per thread
#pragma unroll
        for (int i = 0; i < 2; ++i) {
            const int c = t + i * 256;
            const int row = c >> 3, seg = c & 7;
            g2lds_b128(B + (size_t)(n0 + row) * K + k0 + seg * 4,
                       &Bs[b][row][seg * 4]);
        }
    };

    v8f acc[4];
#pragma unroll
    for (int j = 0; j < 4; ++j) { v8f z = {}; acc[j] = z; }

    copy_chunk(0, 0);
    wait_async();
    __syncthreads();

    const int nchunks = K / BK;
    const int ar = wave * 16 + l16;               // A row owned by this lane
    int buf = 0;
    for (int c = 0; c < nchunks; ++c) {
        if (c + 1 < nchunks) copy_chunk((c + 1) * BK, buf ^ 1);  // prefetch next
#pragma unroll
        for (int kk = 0; kk < BK; kk += 4) {
            const v2f a = *(const v2f*)&As[buf][ar][kk + 2 * half];
#pragma unroll
            for (int j = 0; j < 4; ++j) {
                const v2f b = *(const v2f*)&Bs[buf][j * 16 + l16][kk + 2 * half];
                acc[j] = __builtin_amdgcn_wmma_f32_16x16x4_f32(
                    false, a, false, b, (short)0, acc[j], false, false);
            }
        }
        wait_async();                              // next chunk landed in LDS
        __syncthreads();                           // all waves done with buf
        buf ^= 1;
    }

    // epilogue: C frag reg v -> row = v + 8*(lane/16), col = lane%16
    AS1 float* Cg = (AS1 float*)C;
#pragma unroll
    for (int j = 0; j < 4; ++j) {
        const int col = n0 + j * 16 + l16;
        const float bv = FUSE ? bias[col] : 0.f;
#pragma unroll
        for (int v = 0; v < 8; ++v) {
            const int row = m0 + wave * 16 + v + 8 * half;
            float val = acc[j][v];
            if (FUSE) val = 1.f / (1.f + expf(-(val + bv)));
            Cg[(size_t)row * N + col] = val;
        }
    }
}

// ---------------------------------------------------------------------------
// Sequential rule-creation scan. protos[r] == X[src[r]], SENc[r] == cn2[r]
// always (both branches assign ||x||^2) => stau == base/2 and
// dist(i,r) = SEN[i] + SEN[src[r]] - 2*G[i, src[r]].  Single workgroup.
// ---------------------------------------------------------------------------
__global__ __launch_bounds__(256) void scan_rules(const float* __restrict__ G,
                                                  const float* __restrict__ SEN,
                                                  const float* __restrict__ basep,
                                                  int* __restrict__ src,
                                                  int* __restrict__ rules,
                                                  int* __restrict__ countp, int N) {
    __shared__ int   s_src[MAXN];
    __shared__ float s_sen[MAXN];
    __shared__ float redv[256];
    __shared__ int   redi[256];
    __shared__ int   s_count;
    const int t = threadIdx.x, T = blockDim.x;
    const float stau = 0.5f * (*basep);
    if (t == 0) s_count = 0;
    __syncthreads();

    for (int i = 0; i < N; ++i) {
        const int count = s_count;
        const float sxi = SEN[i];
        const float* Gi = G + (size_t)i * N;
        float best = -INFINITY;
        int bidx = 0x7fffffff;
        for (int r = t; r < count; r += T) {
            const float d = sxi + s_sen[r] - 2.f * Gi[s_src[r]];
            const float dens = expf(-d / stau);
            if (dens > best) { best = dens; bidx = r; }   // keep first max
        }
        redv[t] = best; redi[t] = bidx;
        __syncthreads();
        for (int off = T >> 1; off > 0; off >>= 1) {
            if (t < off) {
                const float bv = redv[t + off]; const int bi = redi[t + off];
                if (bv > redv[t] || (bv == redv[t] && bi < redi[t])) {
                    redv[t] = bv; redi[t] = bi;
                }
            }
            __syncthreads();
        }
        if (t == 0) {
            const float vmax = redv[0];
            const int   idx  = redi[0];
            const int isnew = (count == 0) || (vmax < DELTA_F);
            const int slot  = isnew ? count : idx;
            rules[i] = slot;
            if (isnew) {
                s_src[count] = i; s_sen[count] = sxi; src[count] = i;
                s_count = count + 1;
            }
        }
        __syncthreads();
    }
    if (t == 0) *countp = s_count;
}

// ---------------------------------------------------------------------------
// lamb[n] = dens(n, rules[n]) / sum_r dens(n, r)   using Gram row n.
// ---------------------------------------------------------------------------
__global__ __launch_bounds__(256) void lamb_kernel(const float* __restrict__ G,
                                                   const float* __restrict__ SEN,
                                                   const float* __restrict__ basep,
                                                   const int* __restrict__ src,
                                                   const int* __restrict__ rules,
                                                   const int* __restrict__ countp,
                                                   float* __restrict__ lamb, int N) {
    __shared__ float red[256];
    const int n = blockIdx.x;
    const int count = *countp;
    const float stau = 0.5f * (*basep);
    const float sn = SEN[n];
    const float* Gn = G + (size_t)n * N;
    float s = 0.f;
    for (int r = threadIdx.x; r < count; r += blockDim.x) {
        const int sr = src[r];
        const float d = sn + SEN[sr] - 2.f * Gn[sr];
        s += expf(-d / stau);
    }
    red[threadIdx.x] = s;
    __syncthreads();
    for (int off = blockDim.x >> 1; off > 0; off >>= 1) {
        if (threadIdx.x < off) red[threadIdx.x] += red[threadIdx.x + off];
        __syncthreads();
    }
    if (threadIdx.x == 0) {
        const int sr = src[rules[n]];
        const float d = sn + SEN[sr] - 2.f * Gn[sr];
        lamb[n] = expf(-d / stau) / red[0];
    }
}

// ---------------------------------------------------------------------------
extern "C" void kernel_launch(void* const* d_in, const int* in_sizes, int n_in,
                              void* d_out, int out_size, void* d_ws, size_t ws_size,
                              hipStream_t stream) {
    const float* x  = (const float*)d_in[0];
    const float* W0 = (const float*)d_in[1];
    const float* b0 = (const float*)d_in[2];
    const float* W1 = (const float*)d_in[3];
    const float* b1 = (const float*)d_in[4];

    const int DH   = in_sizes[2];                 // 2048
    const int DOUT = in_sizes[4];                 // 1024
    const int DIN  = in_sizes[1] / DH;            // 2048
    const int N    = in_sizes[0] / DIN;            // 2048

    // workspace layout
    float* G    = (float*)d_ws;                   // N*N
    float* h1   = G + (size_t)N * N;              // N*DH
    float* SEN  = h1 + (size_t)N * DH;            // N
    float* base = SEN + N;                        // 1
    float* gsq  = base + 1;                       // 1
    int* srcA   = (int*)(gsq + 1);                // N
    int* rules  = srcA + N;                       // N
    int* countp = rules + N;                      // 1

    float* out_h2 = (float*)d_out;                // N*DOUT
    float* lamb0  = out_h2 + (size_t)N * DOUT;    // N
    float* lamb1  = lamb0 + N;                    // N

    const dim3 wblk(32, 8);                       // 8 wave32 per block

    // ---------------- layer 0 codebook on x ----------------
    row_sumsq<<<N, 256, 0, stream>>>(x, SEN, DIN);
    zero_scalar<<<1, 1, 0, stream>>>(gsq);
    colmean_sq<<<(DIN + 255) / 256, 256, 0, stream>>>(x, gsq, N, DIN);
    finalize_base<<<1, 256, 0, stream>>>(SEN, gsq, base, N);
    wmma_gemm_xyT<false><<<(N / 128) * (N / 64), wblk, 0, stream>>>(
        x, x, nullptr, G, N, N, DIN);             // G = x x^T
    scan_rules<<<1, 256, 0, stream>>>(G, SEN, base, srcA, rules, countp, N);
    lamb_kernel<<<N, 256, 0, stream>>>(G, SEN, base, srcA, rules, countp, lamb0, N);

    // ---------------- h1 = sigmoid(x W0^T + b0) ----------------
    wmma_gemm_xyT<true><<<(N / 128) * (DH / 64), wblk, 0, stream>>>(
        x, W0, b0, h1, N, DH, DIN);

    // ---------------- layer 1 codebook on h1 ----------------
    row_sumsq<<<N, 256, 0, stream>>>(h1, SEN, DH);
    zero_scalar<<<1, 1, 0, stream>>>(gsq);
    colmean_sq<<<(DH + 255) / 256, 256, 0, stream>>>(h1, gsq, N, DH);
    finalize_base<<<1, 256, 0, stream>>>(SEN, gsq, base, N);
    wmma_gemm_xyT<false><<<(N / 128) * (N / 64), wblk, 0, stream>>>(
        h1, h1, nullptr, G, N, N, DH);            // G = h1 h1^T
    scan_rules<<<1, 256, 0, stream>>>(G, SEN, base, srcA, rules, countp, N);
    lamb_kernel<<<N, 256, 0, stream>>>(G, SEN, base, srcA, rules, countp, lamb1, N);

    // ---------------- h2 = sigmoid(h1 W1^T + b1) ----------------
    wmma_gemm_xyT<true><<<(N / 128) * (DOUT / 64), wblk, 0, stream>>>(
        h1, W1, b1, out_h2, N, DOUT, DH);
}